// SSIM_62234076119826
// MI455X (gfx1250) — compile-verified
//
#include <hip/hip_runtime.h>

typedef __attribute__((ext_vector_type(2))) float v2f;
typedef __attribute__((ext_vector_type(8))) float v8f;

#define IMG_H     512
#define IMG_W     512
#define PLANE_SZ  (IMG_H * IMG_W)
#define NPLANES   48              // 16 batch * 3 channels
#define TILE      16
#define HALO_ROWS 28              // 16 + 2*5 = 26, padded to K multiple of 4
#define HALO_COLS 32              // 26 padded; extra cols get zero weight
#define XPITCH    33              // +1 pad to stagger LDS banks
#define WVPB      4               // waves (tiles) per block
#define TILES_X   (IMG_W / TILE)  // 32
#define TILES_Y   (IMG_H / TILE)  // 32
#define NTILES    (NPLANES * TILES_X * TILES_Y)   // 49152
#define NBLOCKS   (NTILES / WVPB)                 // 12288

#define C1F 0.0001f
#define C2F 0.0009f

__device__ __forceinline__ float fetchv(const float* X1, const float* X2,
                                        int idx, int mode) {
    float a = X1[idx];
    float b = X2[idx];
    switch (mode) {
        case 0:  return a;
        case 1:  return b;
        case 2:  return a * a;
        case 3:  return b * b;
        default: return a * b;
    }
}

__global__ __launch_bounds__(WVPB * 32)
void ssim_tile_wmma(const float* __restrict__ img1,
                    const float* __restrict__ img2,
                    const float* __restrict__ window,
                    float* __restrict__ part_out) {
    __shared__ float sX1[WVPB][HALO_ROWS * XPITCH];
    __shared__ float sX2[WVPB][HALO_ROWS * XPITCH];
    __shared__ float sV [WVPB][TILE * XPITCH];
    __shared__ float sG[11];
    __shared__ float sRed[WVPB * 32];

    const int tid  = threadIdx.x;
    const int wv   = tid >> 5;
    const int lane = tid & 31;
    const int half = lane >> 4;          // 0: lanes 0-15, 1: lanes 16-31
    const int m    = lane & 15;          // row (A) / col (B) index

    // Recover the separable 1-D Gaussian from the 2-D window: k[i] = w[i][5] / sqrt(w[5][5])
    if (tid < 11) sG[tid] = window[tid * 11 + 5] / sqrtf(window[5 * 11 + 5]);
    __syncthreads();

    // Per-lane banded coefficient chunks: serve as A (vertical, G[r,j]=g[j-r])
    // and as B (horizontal, Wh[j,c]=g[j-c]) under the symmetric 16x16x4 layouts.
    v2f aW[7];
#pragma unroll
    for (int q = 0; q < 7; ++q) {
        int j0 = 4 * q + 2 * half;
        int t0 = j0 - m, t1 = j0 + 1 - m;
        float w0 = (t0 >= 0 && t0 <= 10) ? sG[t0] : 0.f;
        float w1 = (t1 >= 0 && t1 <= 10) ? sG[t1] : 0.f;
        aW[q].x = w0;
        aW[q].y = w1;
    }

    // ---- tile coordinates ----
    const int tile_id = blockIdx.x * WVPB + wv;
    const int plane   = tile_id >> 10;                 // / (32*32)
    const int rem     = tile_id & 1023;
    const int r0      = (rem >> 5) * TILE;
    const int c0      = (rem & 31) * TILE;

    const float* p1 = img1 + (size_t)plane * PLANE_SZ;
    const float* p2 = img2 + (size_t)plane * PLANE_SZ;

    float* X1 = sX1[wv];
    float* X2 = sX2[wv];
    float* V  = sV[wv];

    // ---- stage zero-padded halo (28 x 32) into LDS, coalesced by lane ----
    const int gc = c0 - 5 + lane;
    const bool cok = (gc >= 0) && (gc < IMG_W);
#pragma unroll 4
    for (int r = 0; r < HALO_ROWS; ++r) {
        int gr = r0 - 5 + r;
        bool ok = cok && (gr >= 0) && (gr < IMG_H);
        int gidx = gr * IMG_W + gc;
        float a = ok ? p1[gidx] : 0.f;
        float b = ok ? p2[gidx] : 0.f;
        X1[r * XPITCH + lane] = a;
        X2[r * XPITCH + lane] = b;
    }
    __syncthreads();

    // ---- 5 separable convolutions via fp32 WMMA ----
    v8f res[5];
#pragma unroll
    for (int mode = 0; mode < 5; ++mode) {
        // vertical pass: V = G(16x28) x X(28x32), two N-tiles of 16 columns
#pragma unroll
        for (int nt = 0; nt < 2; ++nt) {
            v8f acc = {};
            int colX = nt * 16 + m;
#pragma unroll
            for (int q = 0; q < 7; ++q) {
                int j0 = 4 * q + 2 * half;
                v2f B;
                B.x = fetchv(X1, X2, j0 * XPITCH + colX, mode);
                B.y = fetchv(X1, X2, (j0 + 1) * XPITCH + colX, mode);
                acc = __builtin_amdgcn_wmma_f32_16x16x4_f32(
                    false, aW[q], false, B, (short)0, acc, false, false);
            }
            // C/D layout -> LDS: VGPR v holds row (v + 8*half), col = colX
#pragma unroll
            for (int v = 0; v < 8; ++v)
                V[(v + 8 * half) * XPITCH + colX] = acc[v];
        }
        // same-wave DS ops are in-order (ISA ch.11): V is visible to the reads below

        // horizontal pass: O = V(16x28) x Wh(28x16)
        v8f o = {};
#pragma unroll
        for (int q = 0; q < 7; ++q) {
            int j0 = 4 * q + 2 * half;
            v2f A;
            A.x = V[m * XPITCH + j0];
            A.y = V[m * XPITCH + j0 + 1];
            o = __builtin_amdgcn_wmma_f32_16x16x4_f32(
                false, A, false, aW[q], (short)0, o, false, false);
        }
        res[mode] = o;
    }

    // ---- SSIM map in the C/D register layout + partial sum ----
    float partial = 0.f;
#pragma unroll
    for (int v = 0; v < 8; ++v) {
        float mu1 = res[0][v], mu2 = res[1][v];
        float mu1s = mu1 * mu1;
        float mu2s = mu2 * mu2;
        float m12  = mu1 * mu2;
        float sig1 = res[2][v] - mu1s;
        float sig2 = res[3][v] - mu2s;
        float s12  = res[4][v] - m12;
        float num = (2.f * m12 + C1F) * (2.f * s12 + C2F);
        float den = (mu1s + mu2s + C1F) * (sig1 + sig2 + C2F);
        partial += num / den;
    }

    // ---- deterministic block reduction ----
    sRed[tid] = partial;
    __syncthreads();
#pragma unroll
    for (int s = (WVPB * 32) / 2; s > 0; s >>= 1) {
        if (tid < s) sRed[tid] += sRed[tid + s];
        __syncthreads();
    }
    if (tid == 0) part_out[blockIdx.x] = sRed[0];
}

__global__ __launch_bounds__(256)
void ssim_reduce(const float* __restrict__ part, int n,
                 float* __restrict__ out, float scale) {
    __shared__ float s[256];
    float acc = 0.f;
    for (int i = threadIdx.x; i < n; i += 256) acc += part[i];
    s[threadIdx.x] = acc;
    __syncthreads();
#pragma unroll
    for (int k = 128; k > 0; k >>= 1) {
        if (threadIdx.x < k) s[threadIdx.x] += s[threadIdx.x + k];
        __syncthreads();
    }
    if (threadIdx.x == 0) out[0] = s[0] * scale;
}

extern "C" void kernel_launch(void* const* d_in, const int* in_sizes, int n_in,
                              void* d_out, int out_size, void* d_ws, size_t ws_size,
                              hipStream_t stream) {
    const float* img1   = (const float*)d_in[0];
    const float* img2   = (const float*)d_in[1];
    const float* window = (const float*)d_in[2];
    float* out   = (float*)d_out;
    float* parts = (float*)d_ws;   // NBLOCKS floats = 48 KB scratch

    ssim_tile_wmma<<<NBLOCKS, WVPB * 32, 0, stream>>>(img1, img2, window, parts);

    const float inv_n = 1.0f / (float)(NPLANES * PLANE_SZ);
    ssim_reduce<<<1, 256, 0, stream>>>(parts, NBLOCKS, out, inv_n);
}